// LabelPropagationLoss_27504970563868
// MI455X (gfx1250) — compile-verified
//
#include <hip/hip_runtime.h>
#include <math.h>

typedef __attribute__((ext_vector_type(2))) float v2f;
typedef __attribute__((ext_vector_type(8))) float v8f;

#define NN      8192
#define PP      512
#define KITER   10
#define ALPHA   0.5f

// workspace layout (float offsets); total ~197 KB
#define WS_DINV     0          // [8192]  1/(deg+eps)
#define WS_MSQ_ROW  8192       // [8192]  per-row sum(M^2)
#define WS_E0       16384      // [8192*2] posterior buffer A
#define WS_E1       32768      // [8192*2] posterior buffer B
#define WS_MSQ_TOT  49152      // [1]     mean(M^2)

// ---------------------------------------------------------------------------
// Pass 1 (fused): A_hat = M*(A+I) -> d_out[1:], degrees -> d_inv, sum(M^2) rows.
// 8192 blocks (one row each), 256 threads stride the columns. ~768 MB traffic.
// ---------------------------------------------------------------------------
__global__ void __launch_bounds__(256)
k_ahat_deg(const float* __restrict__ A, const float* __restrict__ M,
           float* __restrict__ Ahat, float* __restrict__ dinv,
           float* __restrict__ msq_row) {
  const int i = blockIdx.x;
  const float* arow = A + (size_t)i * NN;
  const float* mrow = M + (size_t)i * NN;
  float*       orow = Ahat + (size_t)i * NN;
  float deg = 0.0f, msq = 0.0f;
  for (int j = threadIdx.x; j < NN; j += 256) {
    float a = arow[j] + (j == i ? 1.0f : 0.0f);   // A + I (self loop)
    float m = mrow[j];
    orow[j] = m * a;                               // A_hat element (output)
    deg += a;                                      // degree uses A+I, not M*(A+I)
    msq += m * m;
  }
  __shared__ float sd[256], sm[256];
  sd[threadIdx.x] = deg; sm[threadIdx.x] = msq;
  __syncthreads();
  for (int s = 128; s > 0; s >>= 1) {
    if (threadIdx.x < s) { sd[threadIdx.x] += sd[threadIdx.x + s];
                           sm[threadIdx.x] += sm[threadIdx.x + s]; }
    __syncthreads();
  }
  if (threadIdx.x == 0) {
    dinv[i]    = 1.0f / (sd[0] + 1e-8f);
    msq_row[i] = sm[0];
  }
}

// Deterministic reduction of per-row sum(M^2) -> mean(M^2)
__global__ void __launch_bounds__(256)
k_msq_reduce(const float* __restrict__ msq_row, float* __restrict__ out) {
  __shared__ float s[256];
  float acc = 0.0f;
  for (int i = threadIdx.x; i < NN; i += 256) acc += msq_row[i];
  s[threadIdx.x] = acc;
  __syncthreads();
  for (int st = 128; st > 0; st >>= 1) {
    if (threadIdx.x < st) s[threadIdx.x] += s[threadIdx.x + st];
    __syncthreads();
  }
  if (threadIdx.x == 0) out[0] = s[0] / ((float)NN * (float)NN);
}

// E init: fill 0.5 everywhere, then scatter positives (0,1) then negatives (1,0).
__global__ void k_einit(float* __restrict__ E) {
  int idx = blockIdx.x * 256 + threadIdx.x;
  if (idx < 2 * NN) E[idx] = 0.5f;
}
__global__ void __launch_bounds__(512)
k_escatter(const int* __restrict__ pos, const int* __restrict__ neg,
           float* __restrict__ E) {
  int t = threadIdx.x;
  if (t < PP) { int p = pos[t]; E[2*p] = 0.0f; E[2*p+1] = 1.0f; }
  __syncthreads();   // negatives override positives on collision (reference order)
  if (t < PP) { int q = neg[t]; E[2*q] = 1.0f; E[2*q+1] = 0.0f; }
}

// ---------------------------------------------------------------------------
// Propagation step via V_WMMA_F32_16X16X4_F32:
//   Eout = ALPHA*Ein + (1-ALPHA) * d_inv .* (A_hat @ Ein)
// Block = 256 threads = 8 waves; each wave owns 16 rows, loops K by 4.
// E (64 KB) staged in LDS once per block. A operand streamed from global f32.
// EXEC stays all-ones through the WMMA loop (only cndmask on values).
// ---------------------------------------------------------------------------
__global__ void __launch_bounds__(256)
k_prop(const float* __restrict__ Ahat, const float* __restrict__ Ein,
       float* __restrict__ Eout, const float* __restrict__ dinv) {
  __shared__ float sE[2 * NN];               // 64 KB of 320 KB WGP LDS
  for (int i = threadIdx.x; i < 2 * NN; i += 256) sE[i] = Ein[i];  // uniform trip count
  __syncthreads();

  const int lane = threadIdx.x & 31;
  const int wave = threadIdx.x >> 5;
  const int r0   = blockIdx.x * 128 + wave * 16;   // 64 blocks * 8 waves = 512 tiles

  const int  m      = lane & 15;                   // A: M index / D: N index
  const int  half   = lane >> 4;                   // K pair selector
  const int  nc     = (m < 2) ? m : 1;             // clamped B column
  const bool nvalid = (m < 2);                     // only cols 0,1 are real

  // A-fragment source: lane (m,half) reads A_hat[r0+m][k + 2*half + {0,1}]
  const float* arow = Ahat + (size_t)(r0 + m) * NN + 2 * half;

  v8f c = {};
  #pragma unroll 4
  for (int k = 0; k < NN; k += 4) {
    v2f a, b;
    a.x = arow[k];
    a.y = arow[k + 1];
    const int kb = k + 2 * half;                   // B rows K = kb, kb+1 on this lane
    float b0 = sE[kb * 2 + nc];
    float b1 = sE[(kb + 1) * 2 + nc];
    b.x = nvalid ? b0 : 0.0f;                      // zero-pad cols 2..15 (cndmask)
    b.y = nvalid ? b1 : 0.0f;
    c = __builtin_amdgcn_wmma_f32_16x16x4_f32(
        /*neg_a=*/false, a, /*neg_b=*/false, b,
        /*c_mod=*/(short)0, c, /*reuse_a=*/false, /*reuse_b=*/false);
  }

  // D layout: lane holds N=m, rows r0 + v + 8*half for v in 0..7
  if (nvalid) {
    #pragma unroll
    for (int v = 0; v < 8; v++) {
      int row = r0 + v + 8 * half;
      float prop = dinv[row] * c[v];
      float eold = sE[row * 2 + m];
      Eout[row * 2 + m] = ALPHA * eold + (1.0f - ALPHA) * prop;
    }
  }
}

// Final loss: gather, -mean(log(..+eps)), NaN guard, + 0.02*mean(M^2)
__global__ void __launch_bounds__(512)
k_loss(const float* __restrict__ E, const int* __restrict__ pos,
       const int* __restrict__ neg, const float* __restrict__ msq_tot,
       float* __restrict__ out) {
  __shared__ float sp[512], sn[512];
  int t = threadIdx.x;
  float lp = -logf(E[2 * pos[t] + 1] + 1e-10f);
  float ln_ = -logf(E[2 * neg[t] + 0] + 1e-10f);
  sp[t] = lp; sn[t] = ln_;
  __syncthreads();
  for (int s = 256; s > 0; s >>= 1) {
    if (t < s) { sp[t] += sp[t + s]; sn[t] += sn[t + s]; }
    __syncthreads();
  }
  if (t == 0) {
    float pl = sp[0] / (float)PP;
    float nl = sn[0] / (float)PP;
    if (nl != nl) nl = 1.75f;                      // isnan -> 1.75
    out[0] = pl + nl + 0.02f * msq_tot[0];
  }
}

// ---------------------------------------------------------------------------
// Inputs (setup_inputs order): 0 node_embeddings(unused), 1 positive_nodes(int),
// 2 reliable_negatives(int), 3 A(f32), 4 M(f32).
// Output: d_out[0] = total_loss, d_out[1:] = A_hat (8192*8192 f32).
// ---------------------------------------------------------------------------
extern "C" void kernel_launch(void* const* d_in, const int* in_sizes, int n_in,
                              void* d_out, int out_size, void* d_ws, size_t ws_size,
                              hipStream_t stream) {
  const int*   pos = (const int*)d_in[1];
  const int*   neg = (const int*)d_in[2];
  const float* A   = (const float*)d_in[3];
  const float* M   = (const float*)d_in[4];
  float* out  = (float*)d_out;
  float* Ahat = out + 1;                 // 4-byte aligned; all Ahat accesses are b32
  float* ws   = (float*)d_ws;

  float* dinv    = ws + WS_DINV;
  float* msq_row = ws + WS_MSQ_ROW;
  float* E0      = ws + WS_E0;
  float* E1      = ws + WS_E1;
  float* msq_tot = ws + WS_MSQ_TOT;

  k_ahat_deg <<<NN, 256, 0, stream>>>(A, M, Ahat, dinv, msq_row);
  k_msq_reduce<<<1, 256, 0, stream>>>(msq_row, msq_tot);
  k_einit    <<<(2 * NN + 255) / 256, 256, 0, stream>>>(E0);
  k_escatter <<<1, 512, 0, stream>>>(pos, neg, E0);

  float* src = E0; float* dst = E1;
  for (int it = 0; it < KITER; it++) {           // 10 streams over A_hat (min. traffic)
    k_prop<<<NN / 128, 256, 0, stream>>>(Ahat, src, dst, dinv);
    float* tmp = src; src = dst; dst = tmp;
  }
  // KITER even -> final posteriors back in E0 (== src)
  k_loss<<<1, 512, 0, stream>>>(src, pos, neg, msq_tot, out);
}